// CustomLSTM_5274219840177
// MI455X (gfx1250) — compile-verified
//
#include <hip/hip_runtime.h>

typedef float v2f __attribute__((ext_vector_type(2)));
typedef float v8f __attribute__((ext_vector_type(8)));

#define LSTM_B   64
#define LSTM_S   1024
#define LSTM_D   512
#define LSTM_H   512
#define LSTM_G   2048   // 4*H
#define LSTM_NWG 32     // persistent workgroups in phase 2

__device__ __forceinline__ float sigmoid_f(float x) {
  return 1.0f / (1.0f + __expf(-x));
}

__global__ void lstm_init_cnt(unsigned* cnt) { *cnt = 0u; }

// ---------------------------------------------------------------------------
// Phase 1: xg[s*64+b][n] = sum_d x[b,s,d] * Wx[d,n] + bias[n]   (n in [0,2048))
// Each wave computes a 16(M) x 64(N) tile with V_WMMA_F32_16X16X4_F32.
// Grid: 16384 blocks x 256 threads (8 waves) -> 131072 wave-tiles
//       = 4096 M-tiles x 32 N-blocks.
// ---------------------------------------------------------------------------
__global__ __launch_bounds__(256)
void lstm_phase1(const float* __restrict__ x,
                 const float* __restrict__ Wii, const float* __restrict__ Wif,
                 const float* __restrict__ Wig, const float* __restrict__ Wio,
                 const float* __restrict__ bi,  const float* __restrict__ bf,
                 const float* __restrict__ bg,  const float* __restrict__ bo,
                 float* __restrict__ xg)
{
  const int lane  = threadIdx.x & 31;
  const int gwave = blockIdx.x * 8 + (threadIdx.x >> 5);
  const int mtile = gwave >> 5;          // 0..4095
  const int nblk  = gwave & 31;          // 0..31
  const int colbase = nblk * 64;         // never crosses a gate boundary
  const int gate  = colbase >> 9;
  const int nn0   = colbase & 511;
  const int nl    = lane & 15;
  const int c0    = (lane < 16) ? 0 : 2;

  const float* W  = (gate == 0) ? Wii : (gate == 1) ? Wif : (gate == 2) ? Wig : Wio;
  const float* bb = (gate == 0) ? bi  : (gate == 1) ? bf  : (gate == 2) ? bg  : bo;

  // A row r = s*64 + b  (xg is [S,B,4H] row-major -> row index == r)
  const int r = mtile * 16 + nl;
  const int s = r >> 6;
  const int b = r & 63;
  const float* arow = x + ((size_t)b * LSTM_S + s) * LSTM_D;

  v8f a0 = {}, a1 = {}, a2 = {}, a3 = {};
  #pragma unroll 4
  for (int k0 = 0; k0 < LSTM_D; k0 += 4) {
    const float2 af = *(const float2*)(arow + k0 + c0);
    v2f av; av.x = af.x; av.y = af.y;
    const float* w0 = W + (size_t)(k0 + c0) * LSTM_H + nn0 + nl;
    const float* w1 = w0 + LSTM_H;
    v2f b0; b0.x = w0[0];  b0.y = w1[0];
    v2f b1; b1.x = w0[16]; b1.y = w1[16];
    v2f b2; b2.x = w0[32]; b2.y = w1[32];
    v2f b3; b3.x = w0[48]; b3.y = w1[48];
    a0 = __builtin_amdgcn_wmma_f32_16x16x4_f32(false, av, false, b0, (short)0, a0, false, false);
    a1 = __builtin_amdgcn_wmma_f32_16x16x4_f32(false, av, false, b1, (short)0, a1, false, false);
    a2 = __builtin_amdgcn_wmma_f32_16x16x4_f32(false, av, false, b2, (short)0, a2, false, false);
    a3 = __builtin_amdgcn_wmma_f32_16x16x4_f32(false, av, false, b3, (short)0, a3, false, false);
  }

  const float bv0 = bb[nn0 + nl];
  const float bv1 = bb[nn0 + 16 + nl];
  const float bv2 = bb[nn0 + 32 + nl];
  const float bv3 = bb[nn0 + 48 + nl];
  const int hi8 = (lane >= 16) ? 8 : 0;
  #pragma unroll
  for (int i = 0; i < 8; ++i) {
    const size_t row = (size_t)(mtile * 16 + i + hi8);
    float* o = xg + row * LSTM_G + colbase + nl;
    o[0]  = a0[i] + bv0;
    o[16] = a1[i] + bv1;
    o[32] = a2[i] + bv2;
    o[48] = a3[i] + bv3;
  }
}

// ---------------------------------------------------------------------------
// Grid-wide barrier: monotonic counter, agent-scope acq/rel atomics.
// ---------------------------------------------------------------------------
__device__ __forceinline__ void grid_barrier(unsigned* cnt, unsigned target) {
  __syncthreads();
  if (threadIdx.x == 0) {
    __hip_atomic_fetch_add(cnt, 1u, __ATOMIC_ACQ_REL, __HIP_MEMORY_SCOPE_AGENT);
    while (__hip_atomic_load(cnt, __ATOMIC_ACQUIRE, __HIP_MEMORY_SCOPE_AGENT) < target) {
      __builtin_amdgcn_s_sleep(1);
    }
  }
  __syncthreads();
}

// ---------------------------------------------------------------------------
// Phase 2: persistent recurrent scan. 32 WGs x 512 threads (16 waves).
// WG w owns hidden columns [w*16, w*16+16) for all 4 gates.
//   LDS: Wh slab 512x64 (pad 72)   144 KB  (resident all 1024 steps)
//        h stage 64x512 (pad 516)  129 KB  (refilled each step)
//        raw gates 4x64x16          16 KB
//        cell state c 64x16          4 KB
// Wave (g,m): 16x16 WMMA tile rows m*16.., gate g, K=512.
// h double-buffered in global; one grid barrier per step.
// ---------------------------------------------------------------------------
__global__ __launch_bounds__(512)
void lstm_phase2(const float* __restrict__ xg,
                 const float* __restrict__ Whi, const float* __restrict__ Whf,
                 const float* __restrict__ Whg, const float* __restrict__ Who,
                 float* __restrict__ hbuf,          // 2 * 64*512 floats
                 unsigned* __restrict__ cnt,
                 float* __restrict__ out)           // [B,S,H]
{
  __shared__ float sWh[LSTM_D][72];     // [k][g*16+j], stride 72 -> conflict-free B frags
  __shared__ float sH[LSTM_B][516];     // [b][k], stride 516 -> conflict-free A frags
  __shared__ float sG[4][LSTM_B][16];   // raw gate pre-activations
  __shared__ float sC[LSTM_B][16];      // cell state

  const int tid   = threadIdx.x;
  const int lane  = tid & 31;
  const int wave  = tid >> 5;           // 0..15
  const int g     = wave >> 2;          // gate 0..3 (i,f,g,o)
  const int m     = wave & 3;           // batch row tile
  const int j0    = blockIdx.x * 16;    // hidden column base
  const int nl    = lane & 15;
  const int c0    = (lane < 16) ? 0 : 2;
  const int hi8   = (lane >= 16) ? 8 : 0;
  const int rbase = m * 16;

  // Stage the Wh slab once (columns j0..j0+15 of each of the 4 gate blocks).
  for (int e = tid; e < LSTM_D * 64; e += 512) {
    const int k   = e >> 6;
    const int col = e & 63;
    const int gg  = col >> 4;
    const int j   = col & 15;
    const float* wp = (gg == 0) ? Whi : (gg == 1) ? Whf : (gg == 2) ? Whg : Who;
    sWh[k][col] = wp[(size_t)k * LSTM_H + j0 + j];
  }
  // Zero cell state and h buffer 0 slice (h0 = 0).
  for (int e = tid; e < LSTM_B * 16; e += 512) {
    sC[e >> 4][e & 15] = 0.0f;
    hbuf[(size_t)(e >> 4) * LSTM_H + j0 + (e & 15)] = 0.0f;
  }
  grid_barrier(cnt, LSTM_NWG);

  for (int t = 0; t < LSTM_S; ++t) {
    const float* cur = hbuf + (size_t)(t & 1) * (LSTM_B * LSTM_H);
    float*       nxt = hbuf + (size_t)((t + 1) & 1) * (LSTM_B * LSTM_H);

    // Init accumulator tile from xg (independent of h staging).
    v8f acc;
    {
      const size_t colg = (size_t)g * 512 + j0 + nl;
      #pragma unroll
      for (int i = 0; i < 8; ++i) {
        const size_t row = (size_t)t * 64 + (size_t)(rbase + i + hi8);
        acc[i] = xg[row * LSTM_G + colg];
      }
    }

    // Stage h[t] into LDS with 128-bit loads.
    for (int e = tid * 4; e < LSTM_B * LSTM_D; e += 512 * 4) {
      const float4 v = *(const float4*)(cur + e);
      *(float4*)&sH[e >> 9][e & 511] = v;
    }
    __syncthreads();

    // g_tile += h @ Wh   (K=512, fp32 WMMA)
    #pragma unroll 8
    for (int k0 = 0; k0 < LSTM_D; k0 += 4) {
      const float2 af = *(const float2*)&sH[rbase + nl][k0 + c0];
      v2f av; av.x = af.x; av.y = af.y;
      v2f bv; bv.x = sWh[k0 + c0][g * 16 + nl];
              bv.y = sWh[k0 + c0 + 1][g * 16 + nl];
      acc = __builtin_amdgcn_wmma_f32_16x16x4_f32(false, av, false, bv, (short)0, acc, false, false);
    }

    // Publish raw gate values for cross-gate combine.
    #pragma unroll
    for (int i = 0; i < 8; ++i)
      sG[g][rbase + i + hi8][nl] = acc[i];
    __syncthreads();

    // Gate math + cell update: 64x16 elements, 2 per thread.
    for (int e = tid; e < LSTM_B * 16; e += 512) {
      const int b = e >> 4;
      const int j = e & 15;
      const float gi = sigmoid_f(sG[0][b][j]);
      const float gf = sigmoid_f(sG[1][b][j]);
      const float gc = tanhf(sG[2][b][j]);
      const float go = sigmoid_f(sG[3][b][j]);
      const float c  = gf * sC[b][j] + gi * gc;
      sC[b][j] = c;
      const float h = go * tanhf(c);
      nxt[(size_t)b * LSTM_H + j0 + j] = h;
      out[((size_t)b * LSTM_S + t) * LSTM_H + j0 + j] = h;
    }
    grid_barrier(cnt, (unsigned)(t + 2) * LSTM_NWG);
  }
}

// ---------------------------------------------------------------------------
// Launch: workspace layout
//   [0,4)                      barrier counter
//   [1024, 1024+256K)          h ping-pong buffers (2 * 64*512 f32)
//   [1024+256K, +512MB)        xg  [S,B,4H] f32
// ---------------------------------------------------------------------------
extern "C" void kernel_launch(void* const* d_in, const int* in_sizes, int n_in,
                              void* d_out, int out_size, void* d_ws, size_t ws_size,
                              hipStream_t stream) {
  const float* x   = (const float*)d_in[0];
  const float* Wii = (const float*)d_in[1];
  const float* Wif = (const float*)d_in[2];
  const float* Wig = (const float*)d_in[3];
  const float* Wio = (const float*)d_in[4];
  const float* Whi = (const float*)d_in[5];
  const float* Whf = (const float*)d_in[6];
  const float* Whg = (const float*)d_in[7];
  const float* Who = (const float*)d_in[8];
  const float* bi  = (const float*)d_in[9];
  const float* bf  = (const float*)d_in[10];
  const float* bg  = (const float*)d_in[11];
  const float* bo  = (const float*)d_in[12];
  float* out = (float*)d_out;

  char* ws = (char*)d_ws;
  unsigned* cnt = (unsigned*)ws;
  float* hbuf   = (float*)(ws + 1024);
  float* xg     = (float*)(ws + 1024 + (size_t)2 * LSTM_B * LSTM_H * sizeof(float));

  lstm_init_cnt<<<1, 1, 0, stream>>>(cnt);
  lstm_phase1<<<16384, 256, 0, stream>>>(x, Wii, Wif, Wig, Wio, bi, bf, bg, bo, xg);
  lstm_phase2<<<LSTM_NWG, 512, 0, stream>>>(xg, Whi, Whf, Whg, Who, hbuf, cnt, out);
}